// MoEMLP_2224793059576
// MI455X (gfx1250) — compile-verified
//
#include <hip/hip_runtime.h>

// MoE top-1 MLP for MI455X (gfx1250, wave32, WMMA).
// B=4, T=1024, D=1024, E=8, HD=4096.
//
// Roofline: useful work = 68.7 GFLOP fp32; bytes = x 16MB + out 16MB +
// weights 256MB -> HBM floor ~12.4us @ 23.3 TB/s. Memory bound, so:
//  - compute only the routed expert per token (grouped GEMM),
//  - M-block = 32 tokens so each weight fragment read from L2 feeds 2 WMMAs
//    (halves L2 amplification vs M=16 while keeping 128 blocks in flight
//    and 2 workgroups/WGP at ~148.6KB LDS),
//  - fp32 WMMA (V_WMMA_F32_16X16X4_F32) keeps reference precision; matrix
//    throughput is not the limiter.

typedef __attribute__((ext_vector_type(2))) float v2f;
typedef __attribute__((ext_vector_type(8))) float v8f;

#define B_    4
#define T_    1024
#define D_    1024
#define E_    8
#define HD_   4096
#define NTOK  (B_ * T_)          // 4096 tokens

#define MB    32                 // tokens per block (2 WMMA M-subtiles)
#define CHUNK  128               // HD chunk staged through LDS per iteration
#define XPITCH (D_ + 4)          // row step = 4 banks -> conflict-free A reads
#define HPITCH (CHUNK + 4)

// ------------------------------------------------------------------
// Kernel 0: zero the per-expert counters (graph-replay safe re-init)
// ------------------------------------------------------------------
__global__ void moe_zero_counts(int* __restrict__ counts) {
    if (threadIdx.x < E_) counts[threadIdx.x] = 0;
}

// ------------------------------------------------------------------
// Kernel 1: router. One wave32 per token: 8 dot products over D, argmax.
// softmax is monotone -> argmax(logits) == argmax(probs); ascending scan
// with strict '>' reproduces jnp.argmax first-max tie-breaking.
// ------------------------------------------------------------------
__global__ __launch_bounds__(256)
void moe_router(const float* __restrict__ x,
                const float* __restrict__ rw,
                int* __restrict__ expert_id) {
    int gw   = (blockIdx.x * blockDim.x + threadIdx.x) >> 5;  // global wave = token
    int lane = threadIdx.x & 31;
    if (gw >= NTOK) return;

    const float* xr = x + (size_t)gw * D_;
    float acc[E_];
#pragma unroll
    for (int e = 0; e < E_; ++e) acc[e] = 0.0f;

    for (int k = lane; k < D_; k += 32) {
        float xv = xr[k];
#pragma unroll
        for (int e = 0; e < E_; ++e) acc[e] += xv * rw[e * D_ + k];
    }
#pragma unroll
    for (int e = 0; e < E_; ++e) {
#pragma unroll
        for (int off = 16; off >= 1; off >>= 1)
            acc[e] += __shfl_xor(acc[e], off, 32);
    }
    if (lane == 0) {
        int best = 0; float bv = acc[0];
#pragma unroll
        for (int e = 1; e < E_; ++e)
            if (acc[e] > bv) { bv = acc[e]; best = e; }
        expert_id[gw] = best;
    }
}

// ------------------------------------------------------------------
// Kernel 2: scatter token ids into per-expert buckets.
// (bucket order is schedule-dependent, but each token's output row is
// computed from its own data only -> output is deterministic)
// ------------------------------------------------------------------
__global__ void moe_bucket(const int* __restrict__ expert_id,
                           int* __restrict__ counts,
                           int* __restrict__ buckets) {
    int t = blockIdx.x * blockDim.x + threadIdx.x;
    if (t >= NTOK) return;
    int e   = expert_id[t];
    int pos = atomicAdd(&counts[e], 1);
    buckets[e * NTOK + pos] = t;
}

// ------------------------------------------------------------------
// Kernel 3: grouped expert GEMM, fused relu^2.
// grid = (NTOK/MB, E). Block = 256 threads = 8 waves, MB=32 tokens.
//   per HD-chunk (128 wide):
//     GEMM1: wave w -> Hc[32, 16] for h-cols [hc+16w, +16), K = D
//            (each B fragment feeds 2 WMMAs: M-subtiles 0..15 and 16..31)
//     relu^2 -> LDS Hact[32,128]; barrier
//     GEMM2: wave w -> OUT[32, 128 cols at 128w] += Hact @ projT, K = 128
//            (each B fragment feeds 2 WMMAs)
// ------------------------------------------------------------------
__global__ __launch_bounds__(256)
void moe_expert_gemm(const float* __restrict__ x,
                     const float* __restrict__ fc_all,
                     const float* __restrict__ proj_all,
                     const int*  __restrict__ counts,
                     const int*  __restrict__ buckets,
                     float* __restrict__ out) {
    __shared__ float Xs[MB * XPITCH];   // 131.6 KB gathered activations
    __shared__ float Hs[MB * HPITCH];   // 16.9 KB relu^2 staging
    __shared__ int   toks[MB];

    const int e     = blockIdx.y;
    const int mtile = blockIdx.x;
    const int cnt   = counts[e];
    const int row0  = mtile * MB;
    if (row0 >= cnt) return;            // uniform exit: EXEC all-1s for WMMA below

    const int tid  = threadIdx.x;
    const int wave = tid >> 5;
    const int lane = tid & 31;
    const int m    = lane & 15;         // M (or N) index within fragment
    const int kk   = (lane >> 4) << 1;  // 0 for lanes 0-15, 2 for lanes 16-31

    if (tid < MB) {
        int idx  = row0 + tid;
        toks[tid] = (idx < cnt) ? buckets[e * NTOK + idx] : -1;
    }
    __syncthreads();

    // ---- gather MB token rows of x into LDS (wave w loads rows 4w..4w+3) ----
#pragma unroll
    for (int r = 0; r < 4; ++r) {
        int row = wave * 4 + r;
        int tok = toks[row];
        if (tok >= 0) {
            const float4* src = (const float4*)(x + (size_t)tok * D_);
#pragma unroll
            for (int j = 0; j < 8; ++j) {
                float4 v = src[j * 32 + lane];
                float* dst = &Xs[row * XPITCH + (j * 32 + lane) * 4];
                dst[0] = v.x; dst[1] = v.y; dst[2] = v.z; dst[3] = v.w;
            }
        } else {
#pragma unroll
            for (int j = 0; j < 8; ++j) {
                float* dst = &Xs[row * XPITCH + (j * 32 + lane) * 4];
                dst[0] = 0.f; dst[1] = 0.f; dst[2] = 0.f; dst[3] = 0.f;
            }
        }
    }
    __syncthreads();

    const float* fc = fc_all   + (size_t)e * D_ * HD_;   // [D, HD] row-major
    const float* pj = proj_all + (size_t)e * D_ * HD_;   // [D, HD] row-major

    v8f c2[2][8] = {};   // OUT accumulators: 2 M-subtiles x 8 N-tiles = 128 VGPRs

    for (int hc = 0; hc < HD_; hc += CHUNK) {
        // ---------------- GEMM1: Hc = X32 @ fc[:, h0:h0+16] ----------------
        const int h0 = hc + wave * 16;
        v8f c1a = {};            // rows 0..15
        v8f c1b = {};            // rows 16..31
        const float* fcol = fc + h0 + m;   // B fragment N index = m
#pragma unroll 4
        for (int k0 = 0; k0 < D_; k0 += 4) {
            // B: 4x16 f32 frag; rows of fc are HD apart (loaded once, used 2x)
            v2f b;
            b.x = fcol[(size_t)(k0 + kk)     * HD_];
            b.y = fcol[(size_t)(k0 + kk + 1) * HD_];
            // A: 16x4 f32 frags for both M-subtiles
            v2f a0 = *(const v2f*)&Xs[m        * XPITCH + k0 + kk];
            v2f a1 = *(const v2f*)&Xs[(m + 16) * XPITCH + k0 + kk];
            c1a = __builtin_amdgcn_wmma_f32_16x16x4_f32(
                      false, a0, false, b, (short)0, c1a, false, false);
            c1b = __builtin_amdgcn_wmma_f32_16x16x4_f32(
                      false, a1, false, b, (short)0, c1b, false, false);
        }
        // relu^2 and stage into LDS: C layout -> row = r (+8 for hi lanes)
#pragma unroll
        for (int r = 0; r < 8; ++r) {
            int row = (lane < 16) ? r : (r + 8);
            float va = c1a[r];
            va = (va > 0.f) ? va * va : 0.f;
            Hs[row * HPITCH + wave * 16 + m] = va;
            float vb = c1b[r];
            vb = (vb > 0.f) ? vb * vb : 0.f;
            Hs[(row + 16) * HPITCH + wave * 16 + m] = vb;
        }
        __syncthreads();

        // -------- GEMM2: OUT[:, 128w:128w+128] += Hact @ projT[hc:, :] -----
#pragma unroll 2
        for (int k0 = 0; k0 < CHUNK; k0 += 4) {
            v2f a0 = *(const v2f*)&Hs[m        * HPITCH + k0 + kk];
            v2f a1 = *(const v2f*)&Hs[(m + 16) * HPITCH + k0 + kk];
#pragma unroll
            for (int nt = 0; nt < 8; ++nt) {
                int d = wave * 128 + nt * 16 + m;            // B fragment N = d
                // projT[h][d] = pj[d*HD + h]; (h, h+1) contiguous -> b64 load
                v2f b = *(const v2f*)&pj[(size_t)d * HD_ + hc + k0 + kk];
                c2[0][nt] = __builtin_amdgcn_wmma_f32_16x16x4_f32(
                                false, a0, false, b, (short)0, c2[0][nt], false, false);
                c2[1][nt] = __builtin_amdgcn_wmma_f32_16x16x4_f32(
                                false, a1, false, b, (short)0, c2[1][nt], false, false);
            }
        }
        __syncthreads();   // protect Hs before next chunk overwrites it
    }

    // ---- epilogue: scatter OUT rows to their tokens ----
#pragma unroll
    for (int g = 0; g < 2; ++g) {
#pragma unroll
        for (int nt = 0; nt < 8; ++nt) {
#pragma unroll
            for (int r = 0; r < 8; ++r) {
                int row = g * 16 + ((lane < 16) ? r : (r + 8));
                int tok = toks[row];
                if (tok >= 0) {
                    int d = wave * 128 + nt * 16 + m;
                    out[(size_t)tok * D_ + d] = c2[g][nt][r];
                }
            }
        }
    }
}

// ------------------------------------------------------------------
extern "C" void kernel_launch(void* const* d_in, const int* in_sizes, int n_in,
                              void* d_out, int out_size, void* d_ws, size_t ws_size,
                              hipStream_t stream) {
    const float* x   = (const float*)d_in[0];   // [B,T,D]
    const float* rw  = (const float*)d_in[1];   // [E,D]
    const float* fc  = (const float*)d_in[2];   // [E,D,HD]
    const float* pj  = (const float*)d_in[3];   // [E,D,HD]
    float* out = (float*)d_out;                 // [B,T,D]

    int* ws      = (int*)d_ws;
    int* cnts    = ws;              // 8 ints
    int* eid     = ws + 8;          // NTOK ints
    int* buckets = ws + 8 + NTOK;   // E*NTOK ints   (~147 KB total)

    moe_zero_counts<<<1, 32, 0, stream>>>(cnts);
    moe_router<<<NTOK / 8, 256, 0, stream>>>(x, rw, eid);       // 8 waves/block
    moe_bucket<<<NTOK / 256, 256, 0, stream>>>(eid, cnts, buckets);

    dim3 grid(NTOK / MB, E_);       // worst-case M-tiles per expert; empties exit
    moe_expert_gemm<<<grid, 256, 0, stream>>>(x, fc, pj, cnts, buckets, out);
}